// MPS_pytorch_sharing_input_83880711290965
// MI455X (gfx1250) — compile-verified
//
#include <hip/hip_runtime.h>

// MPS scan on MI455X (gfx1250), wave32 + V_WMMA_F32_16X16X4_F32.
//
// Per step l, per sample s:
//   vec_s <- vec_s @ (I + x*t0 + (1-x)*t1) = vec_s @ (C + x_s D),
//   C = I + t1_l, D = t0_l - t1_l.
// Transposed 16x16 state per wave (16 dims x 16 samples):
//   W <- C^T W + x .cols. (D^T W)          (x is per-column == per-lane)
//
// Layout trick: WMMA f32 C/D layout puts rows {r+8h} in VGPR r; the B operand
// of V_WMMA_F32_16X16X4_F32 needs rows {4k+j+2h} in VGPR (k,j). These differ
// by the fixed row permutation sigma = [0,1,4,5,8,9,12,13,2,3,6,7,10,11,14,15]
// (sigma(q+8) = sigma(q)+2). Keeping the state as Y = sigma(W):
//   * Y's 8 VGPRs ARE the four B slices: B_k = {Y[2k], Y[2k+1]} (no shuffles,
//     no selects, no shadow state, zero DS ops);
//   * row-permuting the WMMA output == row-permuting the wave-uniform A
//     operand, i.e. just loading A with column base sigma(lane%16);
//   * un-permute only at the final store: out[sigma(q+8h)] = Y[q].
//
// The identity of C = I + t1 is baked into the packed A operand (not the
// accumulator), so BOTH 4-deep WMMA chains start from the inline constant
// C = 0 -> no per-step Y->accumulator register copies (v_wmma is
// tied-accumulate, so a C=Y start would force 8 VGPR moves per step).
//
// Per step: 8 x V_WMMA_F32_16X16X4_F32 (two 4-deep chains), 4 x b128 A-loads
// (pre-packed {AC,AD} in d_ws, shared by all 1024 waves -> L2/WGP$ resident),
// 1/4 x b128 pixel load, 4 pk_fma. Pure f32 == reference precision (tensors
// are O(1e-9); f16 would flush to zero).

typedef __attribute__((ext_vector_type(2))) float v2f;
typedef __attribute__((ext_vector_type(4))) float v4f;
typedef __attribute__((ext_vector_type(8))) float v8f;

#define L_STEPS 544   // (17-1)*33 + 16 contraction steps
#define PIX     561   // 17*33 pixels per sample (only first 544 used)
#define DD      16    // bond dimension

// sigma(m) = 4*((m>>1)&3) + (m&1) + 2*(m>>3), m in 0..15
__device__ __forceinline__ int sigma_row(int m) {
  return 4 * ((m >> 1) & 3) + (m & 1) + 2 * (m >> 3);
}

// ---------------------------------------------------------------------------
// Prep: pack per-(l,k,lane) A operands {AC.x, AC.y, AD.x, AD.y} as a float4.
//   AC[j] = (I + t1)[l, 4k+j+2h, sigma(lane%16)]   (identity baked in)
//   AD[j] = (t0 - t1)[same]                        (identity cancels)
// pk layout: [l][k][lane][4] floats -> 544*4*32*4*4B = 1,114,112 B in d_ws.
// ---------------------------------------------------------------------------
__global__ __launch_bounds__(256) void mps_pack_kernel(
    const float* __restrict__ tens, float* __restrict__ pk)
{
  const int idx = blockIdx.x * blockDim.x + threadIdx.x;  // (l*4 + k)*32 + lane
  if (idx >= L_STEPS * 4 * 32) return;
  const int lane = idx & 31;
  const int k    = (idx >> 5) & 3;
  const int l    = idx >> 7;
  const int h    = lane >> 4;
  const int colS = sigma_row(lane & 15);

  const int row0 = 4 * k + 0 + 2 * h;
  const int row1 = 4 * k + 1 + 2 * h;
  const float* p0 = tens + ((size_t)(l * DD + row0) * DD + colS) * 2;
  const float* p1 = tens + ((size_t)(l * DD + row1) * DD + colS) * 2;
  const float t00 = p0[0], t10 = p0[1];
  const float t01 = p1[0], t11 = p1[1];

  v4f o;
  o.x = t10 + (row0 == colS ? 1.0f : 0.0f);  // AC.x  (C = I + t1)
  o.y = t11 + (row1 == colS ? 1.0f : 0.0f);  // AC.y
  o.z = t00 - t10;                           // AD.x  (D = t0 - t1)
  o.w = t01 - t11;                           // AD.y
  *(v4f*)(pk + (size_t)idx * 4) = o;
}

// ---------------------------------------------------------------------------
// Main scan kernel. PACKED=true reads pre-packed operands from d_ws,
// PACKED=false reads raw tensors1 (fallback if ws_size is too small).
// ---------------------------------------------------------------------------
template <bool PACKED>
__global__ __launch_bounds__(256) void mps_wmma_kernel(
    const float* __restrict__ xin,    // [16384, 561] pixels (flat [B,N,H,W])
    const float* __restrict__ tsrc,   // packed ws  OR  raw [544,16,16,2]
    float* __restrict__ out)          // [16384, 16]
{
  const int lane   = threadIdx.x & 31;
  const int wave   = (blockIdx.x * blockDim.x + threadIdx.x) >> 5;  // 0..1023
  const int col    = lane & 15;        // sample-in-group == matrix column N
  const int h      = lane >> 4;
  const int sample = wave * DD + col;  // 0..16383

  const float* xp = xin + (size_t)sample * PIX;
  const float* tp = PACKED ? (tsrc + lane * 4)               // pk[.][.][lane][4]
                           : (tsrc + sigma_row(col) * 2);    // raw, col=sigma(M)

  // Fallback path: per-lane diagonal indicators (I of C = I + t1), constant
  // across l, hoisted out of the loop.  dg[k][j] = (4k+j+2h == sigma(col)).
  float dg[4][2];
  if (!PACKED) {
    const int colS = sigma_row(col);
#pragma unroll
    for (int k = 0; k < 4; ++k)
#pragma unroll
      for (int j = 0; j < 2; ++j)
        dg[k][j] = (4 * k + j + 2 * h == colS) ? 1.0f : 0.0f;
  }

  // State Y = sigma-row-permuted W, in WMMA f32 C/D layout.
  // Init W = e0 per column -> Y[0] = 1 on lanes 0-15 (sigma(0)=0), rest 0.
  v8f Y = {};
  Y[0] = (h == 0) ? 1.0f : 0.0f;

  for (int l = 0; l < L_STEPS; ++l) {
    v2f AC[4], AD[4];
    if (PACKED) {
      const float* pl = tp + (size_t)l * (4 * 32 * 4);
#pragma unroll
      for (int k = 0; k < 4; ++k) {
        const v4f v = *(const v4f*)(pl + k * 128);   // 16B aligned
        AC[k].x = v.x; AC[k].y = v.y;
        AD[k].x = v.z; AD[k].y = v.w;
      }
    } else {
      const float* tl = tp + (size_t)l * (DD * DD * 2);
#pragma unroll
      for (int k = 0; k < 4; ++k) {
        const v2f p0 = *(const v2f*)(tl + (4 * k + 0 + 2 * h) * (DD * 2));
        const v2f p1 = *(const v2f*)(tl + (4 * k + 1 + 2 * h) * (DD * 2));
        AC[k].x = p0.y + dg[k][0];  AC[k].y = p1.y + dg[k][1];
        AD[k].x = p0.x - p0.y;      AD[k].y = p1.x - p1.y;
      }
    }

    const float xs = xp[l];  // this column's pixel at step l (same both halves)

    // Two independent 4-deep WMMA accumulate chains, both starting from the
    // inline-constant C = 0 (no state->accumulator copies). B_k taken
    // directly from the state registers (layout identity via sigma).
    v8f a1 = {};
    v8f a2 = {};
#pragma unroll
    for (int k = 0; k < 4; ++k) {
      v2f Bk;
      Bk.x = Y[2 * k];
      Bk.y = Y[2 * k + 1];
      a1 = __builtin_amdgcn_wmma_f32_16x16x4_f32(
          false, AC[k], false, Bk, (short)0, a1, false, false);
      a2 = __builtin_amdgcn_wmma_f32_16x16x4_f32(
          false, AD[k], false, Bk, (short)0, a2, false, false);
    }

    // Y = a1 + x .cols. a2  (per-lane scalar in C/D layout)
#pragma unroll
    for (int r = 0; r < 8; ++r) Y[r] = a1[r] + xs * a2[r];
  }

  // Un-permute on store: Y[q] holds W row sigma(q+8h).
  float* op = out + (size_t)sample * DD;
#pragma unroll
  for (int q = 0; q < 8; ++q) op[sigma_row(q + 8 * h)] = Y[q];
}

extern "C" void kernel_launch(void* const* d_in, const int* in_sizes, int n_in,
                              void* d_out, int out_size, void* d_ws, size_t ws_size,
                              hipStream_t stream) {
  const float* xin  = (const float*)d_in[0];  // input_data [256,64,17,33] f32
  const float* tens = (const float*)d_in[1];  // tensors1   [544,16,16,2] f32
  float* out = (float*)d_out;                 // [256,64,16] f32

  const size_t pack_bytes = (size_t)L_STEPS * 4 * 32 * 4 * sizeof(float); // ~1.1MB

  if (d_ws != nullptr && ws_size >= pack_bytes) {
    float* pk = (float*)d_ws;
    mps_pack_kernel<<<(L_STEPS * 4 * 32 + 255) / 256, 256, 0, stream>>>(tens, pk);
    // 16384 samples / 16 per wave = 1024 waves = 128 blocks x 8 waves.
    mps_wmma_kernel<true><<<128, 256, 0, stream>>>(xin, pk, out);
  } else {
    mps_wmma_kernel<false><<<128, 256, 0, stream>>>(xin, tens, out);
  }
}